// Packets_17617955848819
// MI455X (gfx1250) — compile-verified
//
#include <hip/hip_runtime.h>
#include <math.h>

typedef __attribute__((ext_vector_type(2))) float v2f;
typedef __attribute__((ext_vector_type(8))) float v8f;

#define SEGS_PER_TASK 4   // segments (128 outputs each) per wave; A built once

// sym8 decomposition low-pass (pywt convention), f32
__device__ __constant__ float c_declo[16] = {
    -0.0033824159510061256f, -0.0005421323317911481f, 0.03169508781149298f,
     0.007607487324917605f,  -0.1432942383508097f,   -0.061273359067658524f,
     0.4813596512583722f,     0.7771857517005235f,    0.3644418948353314f,
    -0.05194583810770904f,   -0.027219029917056003f,  0.049137179673607506f,
     0.003808752013890615f,  -0.01495225833704823f,  -0.0003029205147213668f,
     0.0018899503327594609f
};

// reflect-pad read: padded index i maps to source index (i - 14), reflected
__device__ __forceinline__ float load_reflect(const float* __restrict__ x, int L, int i) {
    int m = i - 14;
    m = (m < 0) ? -m : m;
    m = (m >= L) ? (2 * L - 2 - m) : m;
    m = (m < 0) ? 0 : ((m > L - 1) ? (L - 1) : m);   // clamp for overshoot lanes
    return x[m];
}

// One DWT level: in[row][0..L_in) -> out[row*2+c][0..L_out), c in {lo,hi}.
// Per segment: D(16x16) = sum_{t<8} A_t(16x4) x B_t(4x16)
//   A = banded filter matrix (constant), B = im2col window of one node row.
//   D[(c*8+jl), n] = y[c, 8*(n0+n)+jl]
__global__ __launch_bounds__(256)
void wp_level_kernel(const float* __restrict__ in, float* __restrict__ out,
                     int L_in, int L_out, int seg_count, int seg_blocks,
                     int total_tasks, int final_log)
{
    __shared__ float sF[32];            // [c*16 + k] : flipped dec filters
    const int tid = threadIdx.x;
    if (tid < 32) {
        const int c = tid >> 4, k = tid & 15;
        // filt[0][k] = dec_lo[15-k];  filt[1][k] = (k odd ? + : -) dec_lo[k]
        sF[tid] = (c == 0) ? c_declo[15 - k]
                           : ((k & 1) ? c_declo[k] : -c_declo[k]);
    }
    __syncthreads();

    const int lane = tid & 31;
    const int task = blockIdx.x * (blockDim.x >> 5) + (tid >> 5);   // wave-uniform
    if (task >= total_tasks) return;

    const int sb   = task % seg_blocks;
    const int row  = task / seg_blocks;
    const int seg0 = sb * SEGS_PER_TASK;

    const float* __restrict__ xr = in + (size_t)row * (size_t)L_in;

    const int khalf = lane >> 4;        // K-half select (A/B), channel select (D)
    const int mn    = lane & 15;        // M for A, N for B
    const int cch   = mn >> 3;          // A-row channel
    const int jl    = mn & 7;           // A-row local output position

    // ---- Banded filter tiles, built ONCE per wave: A[m,i] = f[c][i-2*jl] ----
    v2f a[8];
#pragma unroll
    for (int t = 0; t < 8; ++t) {
#pragma unroll
        for (int e = 0; e < 2; ++e) {
            const int i   = 4 * t + 2 * khalf + e;
            const int kk  = i - 2 * jl;
            const int idx = (kk < 0) ? 0 : ((kk > 15) ? 15 : kk);
            const float v = sF[cch * 16 + idx];       // ds_load_b32 (dynamic)
            a[t][e] = ((unsigned)kk < 16u) ? v : 0.0f;
        }
    }

    float* __restrict__ orow = out + (size_t)(row * 2 + khalf) * (size_t)L_out;

#pragma unroll 1
    for (int s = 0; s < SEGS_PER_TASK; ++s) {
        const int seg = seg0 + s;
        if (seg >= seg_count) break;                  // wave-uniform

        const int n0 = seg << 4;                      // base 8-wide output block
        const int jb = seg << 7;                      // base output position
        const int w0 = n0 << 4;                       // padded window start (xp index)

        // ---- B tiles: B[i,n] = xp[16*(n0+n) + i] ----
        v2f b[8];
        if (w0 >= 14 && (w0 + 257) < L_in) {
            // interior: no reflection anywhere in the wave's window -> direct loads
            const float* __restrict__ bp = xr + (w0 - 14) + (mn << 4) + 2 * khalf;
#pragma unroll
            for (int t = 0; t < 8; ++t) {
                b[t][0] = bp[4 * t];
                b[t][1] = bp[4 * t + 1];
            }
        } else {
            // edge: reflect-padded gather
            const int ib = w0 + (mn << 4) + 2 * khalf;
#pragma unroll
            for (int t = 0; t < 8; ++t) {
                b[t][0] = load_reflect(xr, L_in, ib + 4 * t);
                b[t][1] = load_reflect(xr, L_in, ib + 4 * t + 1);
            }
        }

        // ---- D = sum_t A_t * B_t (full fp32 WMMA, K accumulated over 8 tiles) ----
        v8f acc = {0.f, 0.f, 0.f, 0.f, 0.f, 0.f, 0.f, 0.f};
#pragma unroll
        for (int t = 0; t < 8; ++t) {
            acc = __builtin_amdgcn_wmma_f32_16x16x4_f32(
                      false, a[t], false, b[t], (short)0, acc, false, false);
        }

        // ---- store: component v -> channel=khalf, j = jb + 8*mn + v ----
        const int j0 = jb + (mn << 3);
        if (jb + 128 <= L_out) {
            if (final_log) {
#pragma unroll
                for (int v = 0; v < 8; ++v) {
                    const float d = acc[v];
                    orow[j0 + v] = logf(fmaf(d, d, 1e-12f));
                }
            } else {
#pragma unroll
                for (int v = 0; v < 8; ++v) orow[j0 + v] = acc[v];
            }
        } else {
            if (final_log) {
#pragma unroll
                for (int v = 0; v < 8; ++v) {
                    const int j = j0 + v;
                    if (j < L_out) {
                        const float d = acc[v];
                        orow[j] = logf(fmaf(d, d, 1e-12f));
                    }
                }
            } else {
#pragma unroll
                for (int v = 0; v < 8; ++v) {
                    const int j = j0 + v;
                    if (j < L_out) orow[j] = acc[v];
                }
            }
        }
    }
}

extern "C" void kernel_launch(void* const* d_in, const int* in_sizes, int n_in,
                              void* d_out, int out_size, void* d_ws, size_t ws_size,
                              hipStream_t stream)
{
    (void)in_sizes; (void)n_in; (void)out_size; (void)ws_size;
    const float* x   = (const float*)d_in[0];     // [128, 65536] f32
    float*       out = (float*)d_out;             // [128, 1, 256, 270] f32
    float*       ws  = (float*)d_ws;

    // Ping-pong intermediates. Max level buffer = 16384 rows x 526 = 8,617,984 floats.
    const size_t STRIDE = 8650752;                // floats, 256-aligned; 2 bufs ~69 MB
    float* bufA = ws;
    float* bufB = ws + STRIDE;

    int L_in = 65536;
    const float* cur_in = x;
    for (int l = 0; l < 8; ++l) {
        const int L_out      = (L_in + 15) >> 1;      // pywt/ptwt dwt length
        const int rows_in    = 128 << l;              // batch * 2^l nodes
        const int seg_count  = (L_out + 127) >> 7;    // 128 outputs per segment
        const int seg_blocks = (seg_count + SEGS_PER_TASK - 1) / SEGS_PER_TASK;
        const int total      = rows_in * seg_blocks;
        const int final_lg   = (l == 7);

        float* cur_out = final_lg ? out : ((l & 1) ? bufB : bufA);

        const int blocks = (total + 7) / 8;           // 8 waves (256 thr) per block
        wp_level_kernel<<<blocks, 256, 0, stream>>>(
            cur_in, cur_out, L_in, L_out, seg_count, seg_blocks, total, final_lg);

        cur_in = cur_out;
        L_in   = L_out;
    }
}